// HieLayer_78288663872088
// MI455X (gfx1250) — compile-verified
//
#include <hip/hip_runtime.h>

typedef __attribute__((ext_vector_type(2))) float v2f;
typedef __attribute__((ext_vector_type(8))) float v8f;
typedef unsigned long long u64t;

static constexpr int Bn = 32;    // batch (docs)
static constexpr int Sn = 2048;  // tokens per doc
static constexpr int En = 512;   // embedding dim
static constexpr int PAD = 8;    // sentinel tokens appended per doc row
static constexpr int TS = Sn + PAD;

// ---------------------------------------------------------------------------
// Prep: per-doc scan of the boundary mask.
//   tok[b*TS + t]  = { lo: sid(t), hi: bitcast(1/len[sid(t)]) }, t < Sn
//   tok[b*TS + t]  = { lo: 0xFFFFFFFF, hi: 0 } sentinel, Sn <= t < TS
//   sstart[b][s]   = first token of sentence s (Sn sentinel for s >= nd)
//   dlens[b]       = number of sentences in doc b
// ---------------------------------------------------------------------------
__global__ __launch_bounds__(256) void hie_prep(const int* __restrict__ bx,
                                                u64t* __restrict__ tok,
                                                int* __restrict__ sstart,
                                                int* __restrict__ dlens) {
  __shared__ int   s_end[Sn];
  __shared__ float s_inv[Sn];
  __shared__ int   s_scan[256];

  const int b   = blockIdx.x;
  const int tid = threadIdx.x;
  const int* row = bx + (size_t)b * Sn;

  for (int s = tid; s < Sn; s += 256) s_inv[s] = 0.0f;

  const int base = tid * 8;  // 8 consecutive tokens per thread
  int flags[8];
  int cnt = 0;
#pragma unroll
  for (int i = 0; i < 8; ++i) { flags[i] = (row[base + i] != 0) ? 1 : 0; cnt += flags[i]; }
  s_scan[tid] = cnt;
  __syncthreads();

  // Hillis-Steele inclusive scan over 256 partials
  for (int off = 1; off < 256; off <<= 1) {
    int v   = s_scan[tid];
    int add = (tid >= off) ? s_scan[tid - off] : 0;
    __syncthreads();
    s_scan[tid] = v + add;
    __syncthreads();
  }
  const int excl = (tid == 0) ? 0 : s_scan[tid - 1];
  const int nd   = s_scan[255];

  int sid[8];
  int run = excl;
#pragma unroll
  for (int i = 0; i < 8; ++i) {
    sid[i] = run;
    if (flags[i]) { s_end[run] = base + i; ++run; }
  }
  __syncthreads();

  for (int s = tid; s < Sn; s += 256) {
    if (s < nd) {
      const int st = (s == 0) ? 0 : (s_end[s - 1] + 1);
      const int ln = s_end[s] - st + 1;
      s_inv[s] = 1.0f / (float)ln;
      sstart[b * (Sn + 1) + s] = st;
    } else {
      sstart[b * (Sn + 1) + s] = Sn;
    }
  }
  if (tid == 0) { sstart[b * (Sn + 1) + Sn] = Sn; dlens[b] = nd; }
  __syncthreads();

#pragma unroll
  for (int i = 0; i < 8; ++i) {
    int sc = sid[i]; if (sc > Sn - 1) sc = Sn - 1;
    const u64t p = ((u64t)__float_as_uint(s_inv[sc]) << 32) | (u64t)(unsigned)sid[i];
    tok[(size_t)b * TS + base + i] = p;
  }
  if (tid < PAD) {  // sentinels: never match any sentence row, weight 0
    tok[(size_t)b * TS + Sn + tid] = 0x00000000FFFFFFFFull;
  }
}

// ---------------------------------------------------------------------------
// Pool: ragged segmented mean as banded f32 matmul via V_WMMA_F32_16X16X4_F32.
// One wave per (doc, 16-sentence group, 64-column strip); 4 accumulators,
// K unrolled 2x (8 tokens / macro-step). tok entries read as indivisible u64
// (prevents conditional partial loads); all addressing is SGPR base + 32-bit
// lane offset. Overshoot tokens get weight 0 via sentinel sid.
// ---------------------------------------------------------------------------
__global__ __launch_bounds__(32) void hie_pool(const float* __restrict__ x,
                                               const u64t* __restrict__ tok,
                                               const int* __restrict__ sstart,
                                               const int dm,
                                               float* __restrict__ out) {
  const int b    = blockIdx.x;
  const int n0   = blockIdx.y * 64;  // 4 tiles x 16 cols
  const int s0   = blockIdx.z * 16;  // 16 sentence slots
  const int lane = threadIdx.x;      // wave32
  const int m    = lane & 15;
  const int kh   = (lane >> 4) << 1; // 0 or 2
  const int tgt  = s0 + m;           // sentence row this lane owns in A
  const int c    = n0 + m;           // lane column offset (elements)

  const int* ss = sstart + b * (Sn + 1);
  const int gstart = ss[s0];
  const int sE     = (s0 + 16 < Sn) ? (s0 + 16) : Sn;
  const int gend   = ss[sE];         // token span of this sentence group

  const float* xbase = x + (size_t)b * Sn * En;   // uniform base, 32-bit offsets
  const u64t*  tb    = tok + (size_t)b * TS;

  v8f a0 = {}, a1 = {}, a2 = {}, a3 = {};

  for (int t0 = gstart; t0 < gend; t0 += 8) {
    const int u0 = t0 + kh;        // tokens for first K=4 chunk (this half-wave)
    const int u1 = u0 + 4;         // tokens for second K=4 chunk

    // Indivisible 64-bit tok loads: {sid, weight} in one b64.
    const u64t qa = tb[u0];
    const u64t qb = tb[u0 + 1];
    const u64t qc = tb[u1];
    const u64t qd = tb[u1 + 1];

    v2f af0, af1;
    af0.x = ((int)(unsigned)qa == tgt) ? __uint_as_float((unsigned)(qa >> 32)) : 0.0f;
    af0.y = ((int)(unsigned)qb == tgt) ? __uint_as_float((unsigned)(qb >> 32)) : 0.0f;
    af1.x = ((int)(unsigned)qc == tgt) ? __uint_as_float((unsigned)(qc >> 32)) : 0.0f;
    af1.y = ((int)(unsigned)qd == tgt) ? __uint_as_float((unsigned)(qd >> 32)) : 0.0f;

    // B rows: clamp row index (weights are 0 for clamped rows); 32-bit offsets.
    const int r0 = (u0     < Sn) ? u0     : (Sn - 1);
    const int r1 = (u0 + 1 < Sn) ? u0 + 1 : (Sn - 1);
    const int r2 = (u1     < Sn) ? u1     : (Sn - 1);
    const int r3 = (u1 + 1 < Sn) ? u1 + 1 : (Sn - 1);
    const int o0 = r0 * En + c;
    const int o1 = r1 * En + c;
    const int o2 = r2 * En + c;
    const int o3 = r3 * En + c;

    v2f b00, b01, b02, b03, b10, b11, b12, b13;
    b00.x = xbase[o0];      b00.y = xbase[o1];
    b01.x = xbase[o0 + 16]; b01.y = xbase[o1 + 16];
    b02.x = xbase[o0 + 32]; b02.y = xbase[o1 + 32];
    b03.x = xbase[o0 + 48]; b03.y = xbase[o1 + 48];
    b10.x = xbase[o2];      b10.y = xbase[o3];
    b11.x = xbase[o2 + 16]; b11.y = xbase[o3 + 16];
    b12.x = xbase[o2 + 32]; b12.y = xbase[o3 + 32];
    b13.x = xbase[o2 + 48]; b13.y = xbase[o3 + 48];

    a0 = __builtin_amdgcn_wmma_f32_16x16x4_f32(false, af0, false, b00, (short)0, a0, false, false);
    a1 = __builtin_amdgcn_wmma_f32_16x16x4_f32(false, af0, false, b01, (short)0, a1, false, false);
    a2 = __builtin_amdgcn_wmma_f32_16x16x4_f32(false, af0, false, b02, (short)0, a2, false, false);
    a3 = __builtin_amdgcn_wmma_f32_16x16x4_f32(false, af0, false, b03, (short)0, a3, false, false);
    a0 = __builtin_amdgcn_wmma_f32_16x16x4_f32(false, af1, false, b10, (short)0, a0, false, false);
    a1 = __builtin_amdgcn_wmma_f32_16x16x4_f32(false, af1, false, b11, (short)0, a1, false, false);
    a2 = __builtin_amdgcn_wmma_f32_16x16x4_f32(false, af1, false, b12, (short)0, a2, false, false);
    a3 = __builtin_amdgcn_wmma_f32_16x16x4_f32(false, af1, false, b13, (short)0, a3, false, false);
  }

  // Store 16x64 tile; empty slots hold exact zeros -> implements padding.
  float* obase = out + (size_t)b * dm * En;
  const int rbase = (lane < 16) ? 0 : 8;
#pragma unroll
  for (int r = 0; r < 8; ++r) {
    const int slot = s0 + r + rbase;
    if (slot < dm) {
      const int oo = slot * En + c;
      obase[oo]      = a0[r];
      obase[oo + 16] = a1[r];
      obase[oo + 32] = a2[r];
      obase[oo + 48] = a3[r];
    }
  }
}

// ---------------------------------------------------------------------------
// Tail: doc_lens as floats after the doc_out block.
// ---------------------------------------------------------------------------
__global__ void hie_lens(const int* __restrict__ dlens, float* __restrict__ out,
                         const size_t basei, const int tail) {
  const int i = blockIdx.x * blockDim.x + threadIdx.x;
  if (i < tail) out[basei + i] = (i < Bn) ? (float)dlens[i] : 0.0f;
}

extern "C" void kernel_launch(void* const* d_in, const int* in_sizes, int n_in,
                              void* d_out, int out_size, void* d_ws, size_t ws_size,
                              hipStream_t stream) {
  (void)in_sizes; (void)n_in; (void)ws_size;
  const float* x  = (const float*)d_in[0];
  const int*   bx = (const int*)d_in[1];
  float* out = (float*)d_out;

  // Workspace carve: tok (526KB) | sstart (262KB) | dlens (128B)
  u64t* tok    = (u64t*)d_ws;
  int* sstart  = (int*)(tok + (size_t)Bn * TS);
  int* dlens   = sstart + Bn * (Sn + 1);

  // out_size = Bn*dm*En (+ Bn tail for doc_lens)
  const int dm   = out_size / (Bn * En);
  const int tail = out_size - dm * (Bn * En);

  hie_prep<<<Bn, 256, 0, stream>>>(bx, tok, sstart, dlens);

  dim3 grid(Bn, En / 64, (dm + 15) / 16);
  hie_pool<<<grid, 32, 0, stream>>>(x, tok, sstart, dm, out);

  if (tail > 0) {
    hie_lens<<<(tail + 255) / 256, 256, 0, stream>>>(dlens, out,
                                                     (size_t)Bn * dm * En, tail);
  }
}